// MahalanobisClassifier_68247030333740
// MI455X (gfx1250) — compile-verified
//
#include <hip/hip_runtime.h>
#include <stdint.h>

// Mahalanobis classifier as 2-plane WMMA GEMM:
//   out[b,c] = sum_d x2[b,d]*(-w[c,d]) + x[b,d]*(2 p w)[c,d]  -  bias[c]
// with w = exp(log_diag), bias[c] = sum_d p^2 w.
//
// GEMM uses f32 WMMA (V_WMMA_F32_16X16X4_F32) with double-buffered LDS fed by
// CDNA5 async global->LDS loads (GLOBAL_LOAD_ASYNC_TO_LDS_B128 / ASYNCcnt).

typedef __attribute__((ext_vector_type(2))) float v2f;
typedef __attribute__((ext_vector_type(8))) float v8f;

#define BM    2048   // batch (GEMM M)
#define CN    1000   // classes (GEMM N)
#define CNPAD 1024   // padded N so 128-wide tiles need no load masking
#define DK    256    // features (GEMM K)

#define LDSS  20          // LDS row stride in floats (80B: 16B-aligned b128,
                          // 8B-aligned b64 frag loads, conflict-free banks)
#define TILEF (128 * LDSS) // floats per staged tile

// ---- CDNA5 async global->LDS (ASYNCcnt-tracked), GV addressing mode --------
__device__ __forceinline__ void async_load_b128(uint32_t lds_addr,
                                                const float* gaddr) {
  asm volatile("global_load_async_to_lds_b128 %0, %1, off"
               :: "v"(lds_addr), "v"(gaddr)
               : "memory");
}
__device__ __forceinline__ void wait_async_lds() {
  asm volatile("s_wait_asynccnt 0x0" ::: "memory");
}

// ---------------------------------------------------------------------------
// Prep: build GEMM B-planes (padded to CNPAD rows, zero-filled) + bias vector.
// ---------------------------------------------------------------------------
__global__ __launch_bounds__(256)
void maha_prep_kernel(const float* __restrict__ proto,
                      const float* __restrict__ logd,
                      float* __restrict__ wneg,   // [CNPAD][DK]  = -exp(ld)
                      float* __restrict__ pw2,    // [CNPAD][DK]  = 2*p*exp(ld)
                      float* __restrict__ bias) { // [CNPAD]      = sum p^2 exp(ld)
  __shared__ float red[256];
  const int c = blockIdx.x;
  const int d = threadIdx.x;

  float wn = 0.0f, pw = 0.0f, p2w = 0.0f;
  if (c < CN) {
    const int idx = c * DK + d;
    const float w = expf(logd[idx]);
    const float p = proto[idx];
    wn  = -w;
    pw  = 2.0f * p * w;
    p2w = p * p * w;
  }
  const int oidx = c * DK + d;
  wneg[oidx] = wn;
  pw2[oidx]  = pw;

  red[d] = p2w;
  __syncthreads();
  #pragma unroll
  for (int s = 128; s > 0; s >>= 1) {
    if (d < s) red[d] += red[d + s];
    __syncthreads();
  }
  if (d == 0) bias[c] = red[0];
}

// ---------------------------------------------------------------------------
// WMMA GEMM: grid (CNPAD/128, BM/128), 256 threads = 8 waves per block.
// Block tile 128x128; wave tile 64x32 = 4x2 fragments of 16x16 f32.
// K in chunks of 16; double-buffered LDS fed by async b128 loads.
// ---------------------------------------------------------------------------
__global__ __launch_bounds__(256)
void maha_gemm_kernel(const float* __restrict__ x,
                      const float* __restrict__ wneg,
                      const float* __restrict__ pw2,
                      const float* __restrict__ bias,
                      float* __restrict__ out) {
  // smem[buf][arr][tile] : buf = 0/1, arr = {x, -w, 2pw}
  __shared__ __align__(16) float smem[2 * 3 * TILEF];

  const int tid    = threadIdx.x;
  const int lane   = tid & 31;
  const int wave   = tid >> 5;
  const int wave_m = wave & 1;   // 2 row-halves of 64
  const int wave_n = wave >> 1;  // 4 col-quarters of 32

  const int m0 = blockIdx.y * 128;
  const int n0 = blockIdx.x * 128;

  const int mlane = lane & 15;        // M (or N) index within a fragment
  const int khalf = (lane >> 4) * 2;  // K sub-offset per A/B layout (0 or 2)

  // staging geometry: thread owns b128 segments (row0,c4) and (row1,c4)
  const int row0 = tid >> 2;          // 0..63
  const int row1 = row0 + 64;         // 64..127
  const int c4   = (tid & 3) * 4;     // 0,4,8,12

  const uint32_t lds_base = (uint32_t)(uintptr_t)smem;
  const uint32_t st0 = (uint32_t)((row0 * LDSS + c4) * 4);
  const uint32_t st1 = (uint32_t)((row1 * LDSS + c4) * 4);

  const float* ga_x0 = x    + (m0 + row0) * DK + c4;
  const float* ga_x1 = x    + (m0 + row1) * DK + c4;
  const float* ga_w0 = wneg + (n0 + row0) * DK + c4;
  const float* ga_w1 = wneg + (n0 + row1) * DK + c4;
  const float* ga_p0 = pw2  + (n0 + row0) * DK + c4;
  const float* ga_p1 = pw2  + (n0 + row1) * DK + c4;

  v8f acc[4][2];
  #pragma unroll
  for (int fm = 0; fm < 4; ++fm)
    #pragma unroll
    for (int fn = 0; fn < 2; ++fn)
      acc[fm][fn] = (v8f){0.f, 0.f, 0.f, 0.f, 0.f, 0.f, 0.f, 0.f};

  auto stage = [&](int buf, int kc) {
    const uint32_t b = lds_base + (uint32_t)(buf * 3 * TILEF * 4);
    async_load_b128(b + 0u * TILEF * 4u + st0, ga_x0 + kc);
    async_load_b128(b + 0u * TILEF * 4u + st1, ga_x1 + kc);
    async_load_b128(b + 1u * TILEF * 4u + st0, ga_w0 + kc);
    async_load_b128(b + 1u * TILEF * 4u + st1, ga_w1 + kc);
    async_load_b128(b + 2u * TILEF * 4u + st0, ga_p0 + kc);
    async_load_b128(b + 2u * TILEF * 4u + st1, ga_p1 + kc);
  };

  // prologue: stage chunk 0 into buffer 0
  stage(0, 0);
  wait_async_lds();
  __syncthreads();

  const int NCHUNK = DK / 16;  // 16
  for (int i = 0; i < NCHUNK; ++i) {
    // issue async loads for next chunk into the other buffer; they overlap
    // with this chunk's WMMA chain (target buffer's readers all passed the
    // barrier that ended iteration i-1).
    if (i + 1 < NCHUNK) stage((i + 1) & 1, (i + 1) * 16);

    const float* xs  = smem + (i & 1) * 3 * TILEF;
    const float* bsw = xs + TILEF;
    const float* bsp = bsw + TILEF;

    #pragma unroll
    for (int k4 = 0; k4 < 4; ++k4) {
      const int kb = k4 * 4 + khalf;

      v2f afrag[4], a2frag[4];
      #pragma unroll
      for (int fm = 0; fm < 4; ++fm) {
        const int r = wave_m * 64 + fm * 16 + mlane;
        afrag[fm]  = *(const v2f*)(xs + r * LDSS + kb);
        a2frag[fm] = afrag[fm] * afrag[fm];   // x^2 fragment, same layout
      }
      v2f bw[2], bp[2];
      #pragma unroll
      for (int fn = 0; fn < 2; ++fn) {
        const int r = wave_n * 32 + fn * 16 + mlane;
        bw[fn] = *(const v2f*)(bsw + r * LDSS + kb);
        bp[fn] = *(const v2f*)(bsp + r * LDSS + kb);
      }

      #pragma unroll
      for (int fm = 0; fm < 4; ++fm)
        #pragma unroll
        for (int fn = 0; fn < 2; ++fn) {
          acc[fm][fn] = __builtin_amdgcn_wmma_f32_16x16x4_f32(
              false, a2frag[fm], false, bw[fn], (short)0, acc[fm][fn],
              false, false);
          acc[fm][fn] = __builtin_amdgcn_wmma_f32_16x16x4_f32(
              false, afrag[fm], false, bp[fn], (short)0, acc[fm][fn],
              false, false);
        }
    }

    wait_async_lds();   // next chunk's tiles fully landed in LDS
    __syncthreads();    // ... and visible to all waves
  }

  // --- epilogue: C layout is VGPR r -> M=(lane>=16)*8+r, N=lane&15 ---
  #pragma unroll
  for (int fm = 0; fm < 4; ++fm) {
    #pragma unroll
    for (int fn = 0; fn < 2; ++fn) {
      const int ncol = n0 + wave_n * 32 + fn * 16 + mlane;
      if (ncol < CN) {
        const float bv = bias[ncol];
        const int mbase = m0 + wave_m * 64 + fm * 16 + (lane >> 4) * 8;
        #pragma unroll
        for (int r = 0; r < 8; ++r) {
          out[(mbase + r) * CN + ncol] = acc[fm][fn][r] - bv;
        }
      }
    }
  }
}

// ---------------------------------------------------------------------------
extern "C" void kernel_launch(void* const* d_in, const int* in_sizes, int n_in,
                              void* d_out, int out_size, void* d_ws, size_t ws_size,
                              hipStream_t stream) {
  const float* x     = (const float*)d_in[0];  // [B, D]
  const float* proto = (const float*)d_in[1];  // [C, D]
  const float* logd  = (const float*)d_in[2];  // [C, D]
  float* out = (float*)d_out;                  // [B, C]

  // workspace: wneg [CNPAD*DK] | pw2 [CNPAD*DK] | bias [CNPAD]  (~2.1 MB)
  float* wneg = (float*)d_ws;
  float* pw2  = wneg + CNPAD * DK;
  float* bias = pw2 + CNPAD * DK;

  maha_prep_kernel<<<CNPAD, 256, 0, stream>>>(proto, logd, wneg, pw2, bias);
  maha_gemm_kernel<<<dim3(CNPAD / 128, BM / 128), 256, 0, stream>>>(
      x, wneg, pw2, bias, out);
}